// Criterion_28278064676994
// MI455X (gfx1250) — compile-verified
//
#include <hip/hip_runtime.h>
#include <math.h>

#define MARGIN_F 0.2f
#define EPS_F    1e-8f

typedef float v2f __attribute__((ext_vector_type(2)));
typedef float v8f __attribute__((ext_vector_type(8)));

// One wave handles 16 triplets per "group". Squared distances computed as the
// diagonal of D*D^T via V_WMMA_F32_16X16X4_F32 accumulation over K=128.
__global__ void triplet_margin_wmma_kernel(const float* __restrict__ batch,
                                           const float* __restrict__ beta,
                                           const int*   __restrict__ labels,
                                           const int*   __restrict__ triplets,
                                           float*       __restrict__ acc,   // acc[0]=total, acc[1]=count
                                           int T, int groups)
{
    const int lane          = threadIdx.x & 31;
    const int waveInBlock   = threadIdx.x >> 5;
    const int wavesPerBlock = blockDim.x >> 5;
    const int wave          = blockIdx.x * wavesPerBlock + waveInBlock;
    const int waveStride    = gridDim.x * wavesPerBlock;

    const int trip    = lane & 15;            // triplet slot (duplicated across wave halves)
    const int colBase = (lane >> 4) << 1;     // lanes 0-15 -> K {0,1}; lanes 16-31 -> K {2,3}

    float localTot = 0.f, localCnt = 0.f;

    for (int g = wave; g < groups; g += waveStride) {
        const int tripIdx = g * 16 + trip;
        const float valid = (tripIdx < T && lane < 16) ? 1.f : 0.f;
        const int   ti    = (tripIdx < T) ? tripIdx : (T - 1);   // clamp: loads stay in bounds

        const int t0 = triplets[ti * 3 + 0];
        const int t1 = triplets[ti * 3 + 1];
        const int t2 = triplets[ti * 3 + 2];

        const float* rowA = batch + (size_t)t0 * 128 + colBase;
        const float* rowP = batch + (size_t)t1 * 128 + colBase;
        const float* rowN = batch + (size_t)t2 * 128 + colBase;

        v8f cap = {};   // accumulates Dap * Dap^T  (16x16)
        v8f can = {};   // accumulates Dan * Dan^T  (16x16)

        #pragma unroll
        for (int k = 0; k < 32; ++k) {                 // 32 chunks of K=4
            v2f a = *(const v2f*)(rowA + 4 * k);
            v2f p = *(const v2f*)(rowP + 4 * k);
            v2f n = *(const v2f*)(rowN + 4 * k);
            v2f dap = a - p;
            v2f dan = a - n;
            // A and B layouts coincide for the symmetric Gram product: pass dap twice.
            cap = __builtin_amdgcn_wmma_f32_16x16x4_f32(false, dap, false, dap,
                                                        (short)0, cap, false, false);
            can = __builtin_amdgcn_wmma_f32_16x16x4_f32(false, dan, false, dan,
                                                        (short)0, can, false, false);
        }

        // Diagonal element (i,i) of C lives in VGPR (i&7), lane (i<8 ? i : i+16).
        const int slane = (trip < 8) ? trip : (trip + 16);
        const int vsel  = trip & 7;
        float d2ap = 0.f, d2an = 0.f;
        #pragma unroll
        for (int v = 0; v < 8; ++v) {
            float tap = __shfl(cap[v], slane, 32);
            float tnn = __shfl(can[v], slane, 32);
            d2ap += (vsel == v) ? tap : 0.f;
            d2an += (vsel == v) ? tnn : 0.f;
        }

        const float dap = sqrtf(d2ap + EPS_F);
        const float dan = sqrtf(d2an + EPS_F);
        const float b   = beta[labels[t0]];
        const float pos = fmaxf(dap - b + MARGIN_F, 0.f);
        const float neg = fmaxf(b - dan + MARGIN_F, 0.f);

        localTot += valid * (pos + neg);
        localCnt += valid * ((pos > 0.f || neg > 0.f) ? 1.f : 0.f);
    }

    // wave32 tree reduction
    #pragma unroll
    for (int off = 16; off >= 1; off >>= 1) {
        localTot += __shfl_xor(localTot, off, 32);
        localCnt += __shfl_xor(localCnt, off, 32);
    }

    __shared__ float sTot[32];
    __shared__ float sCnt[32];
    if (lane == 0) { sTot[waveInBlock] = localTot; sCnt[waveInBlock] = localCnt; }
    __syncthreads();
    if (threadIdx.x == 0) {
        float t = 0.f, c = 0.f;
        for (int w = 0; w < wavesPerBlock; ++w) { t += sTot[w]; c += sCnt[w]; }
        atomicAdd(&acc[0], t);
        atomicAdd(&acc[1], c);
    }
}

__global__ void triplet_margin_finalize(const float* __restrict__ acc,
                                        float* __restrict__ out)
{
    const float tot = acc[0];
    const float cnt = acc[1];
    out[0] = (cnt == 0.f) ? tot : (tot / fmaxf(cnt, 1.f));
}

extern "C" void kernel_launch(void* const* d_in, const int* in_sizes, int n_in,
                              void* d_out, int out_size, void* d_ws, size_t ws_size,
                              hipStream_t stream) {
    const float* batch    = (const float*)d_in[0];   // (16384, 128) f32
    const float* beta     = (const float*)d_in[1];   // (1000,)      f32
    const int*   labels   = (const int*)  d_in[2];   // (16384,)     i32
    const int*   triplets = (const int*)  d_in[3];   // (T, 3)       i32

    const int T      = in_sizes[3] / 3;
    const int groups = (T + 15) / 16;

    float* acc = (float*)d_ws;
    hipMemsetAsync(acc, 0, 2 * sizeof(float), stream);

    const int threads       = 256;                   // 8 waves / block
    const int wavesPerBlock = threads / 32;
    int blocks = (groups + wavesPerBlock - 1) / wavesPerBlock;
    if (blocks > 32768) blocks = 32768;              // grid-stride covers the rest

    triplet_margin_wmma_kernel<<<blocks, threads, 0, stream>>>(
        batch, beta, labels, triplets, acc, T, groups);
    triplet_margin_finalize<<<1, 1, 0, stream>>>(acc, (float*)d_out);
}